// Voroloss_semi_opt_81286551044465
// MI455X (gfx1250) — compile-verified
//
#include <hip/hip_runtime.h>
#include <hip/hip_bf16.h>

typedef __attribute__((ext_vector_type(2))) float v2f;
typedef __attribute__((ext_vector_type(8))) float v8f;

#define NPTS 16384
#define MSP  4096
#define KNN  11
#define BLK  256   // 8 waves of 32
#define BPB  64    // blocks per batch (16384 / 256)

__device__ __forceinline__ unsigned float_to_ordered_u32(float f) {
    unsigned u = __float_as_uint(f);
    // monotone map: f1 < f2  <=>  map(f1) < map(f2) (unsigned)
    return u ^ (((unsigned)((int)u >> 31)) | 0x80000000u);
}

__global__ __launch_bounds__(BLK) void voroloss_kernel(
    const float* __restrict__ points,
    const float* __restrict__ spoints,
    float* __restrict__ out)
{
    // 48KB spoints SoA + 16KB per-wave C-tile transpose scratch = 64KB LDS
    __shared__ float sxs[MSP];
    __shared__ float sys[MSP];
    __shared__ float szs[MSP];
    __shared__ __align__(16) float ws[8 * 32 * 16];

    const int tid  = threadIdx.x;
    const int lane = tid & 31;
    const int wave = tid >> 5;
    const int b    = blockIdx.x >> 6;          // batch index (blocks/batch = 64)
    const int pb0  = (blockIdx.x & (BPB - 1)) * BLK;

    const float* __restrict__ P = points  + (size_t)b * NPTS * 3;
    const float* __restrict__ S = spoints + (size_t)b * MSP  * 3;

    // Stage spoints into LDS (SoA). L2-resident; each batch's 64 blocks re-read 48KB.
    for (int i = tid; i < MSP; i += BLK) {
        float x = S[i * 3 + 0];
        float y = S[i * 3 + 1];
        float z = S[i * 3 + 2];
        sxs[i] = x; sys[i] = y; szs[i] = z;
    }
    __syncthreads();

    const int  col = lane & 15;        // N-column this lane supplies/consumes
    const bool hi  = (lane >> 4) != 0; // half-wave: false => K=0,1 ; true => K=2,3

    // A-matrix (16x4 f32, loop-invariant): row M = col of points.
    // A = [-2px, -2py, -2pz, 1]  so  A*B with B=[sx;sy;sz;|s|^2] = |s|^2 - 2 p.s
    v2f a0, a1;
    {
        int r0 = pb0 + wave * 32 + col;   // tile 0 rows: points 0..15 of this wave
        int r1 = r0 + 16;                 // tile 1 rows: points 16..31
        float x0 = P[r0 * 3 + 0], y0 = P[r0 * 3 + 1], z0 = P[r0 * 3 + 2];
        float x1 = P[r1 * 3 + 0], y1 = P[r1 * 3 + 1], z1 = P[r1 * 3 + 2];
        a0.x = hi ? (-2.f * z0) : (-2.f * x0);
        a0.y = hi ? 1.0f        : (-2.f * y0);
        a1.x = hi ? (-2.f * z1) : (-2.f * x1);
        a1.y = hi ? 1.0f        : (-2.f * y1);
    }

    // Top-KNN as sorted (ascending) order-preserving u32 keys + indices.
    unsigned ka[KNN];
    unsigned ia[KNN];
#pragma unroll
    for (int j = 0; j < KNN; ++j) { ka[j] = 0xFFFFFFFFu; ia[j] = 0u; }

    float* wbase = ws + wave * 512;                 // 32 rows x 16 cols
    float* wrow  = wbase + (hi ? 128 : 0) + col;    // store base: row (8*hi + r), col
    const float* rrow = wbase + lane * 16;          // read base: this lane's point row

    for (int nt = 0; nt < MSP / 16; ++nt) {
        // B-matrix (4x16 f32): column N = nt*16 + col. Branch-free lane select.
        int n = nt * 16 + col;
        float sx = sxs[n], sy = sys[n], sz = szs[n];
        float snorm = sx * sx + sy * sy + sz * sz;
        v2f bm;
        bm.x = hi ? sz    : sx;
        bm.y = hi ? snorm : sy;

        v8f c0 = {};
        v8f c1 = {};
        c0 = __builtin_amdgcn_wmma_f32_16x16x4_f32(false, a0, false, bm, (short)0, c0, false, false);
        c1 = __builtin_amdgcn_wmma_f32_16x16x4_f32(false, a1, false, bm, (short)0, c1, false, false);

        // Transpose C tiles through wave-private LDS: point-major rows.
        // c0[r] is key for point (r + 8*hi), column col ; c1[r] for point 16 + r + 8*hi.
#pragma unroll
        for (int r = 0; r < 8; ++r) {
            wrow[r * 16]       = c0[r];
            wrow[256 + r * 16] = c1[r];
        }
        asm volatile("s_wait_dscnt 0" ::: "memory");   // wave-local LDS RAW ordering

        // Each lane scans the 16 keys of its own point.
        float key[16];
#pragma unroll
        for (int q = 0; q < 4; ++q) {
            float4 v = ((const float4*)rrow)[q];
            key[q * 4 + 0] = v.x; key[q * 4 + 1] = v.y;
            key[q * 4 + 2] = v.z; key[q * 4 + 3] = v.w;
        }
#pragma unroll
        for (int k = 0; k < 16; ++k) {
            unsigned kv = float_to_ordered_u32(key[k]);
            if (kv < ka[KNN - 1]) {          // single cheap guard branch, rare after warmup
                unsigned iv = (unsigned)(nt * 16 + k);
                // Branchless compare-swap bubble: pure bitwise, nothing for the
                // backend to turn into exec-mask diamonds. Keeps array sorted.
#pragma unroll
                for (int j = 0; j < KNN; ++j) {
                    unsigned m  = 0u - (unsigned)(kv < ka[j]);  // all-ones if new key smaller
                    unsigned dk = (ka[j] ^ kv) & m;
                    unsigned di = (ia[j] ^ iv) & m;
                    ka[j] ^= dk; kv ^= dk;
                    ia[j] ^= di; iv ^= di;
                }
            }
        }
    }

    // Voronoi-edge loss from the 11 neighbors (all coords still in LDS).
    int pid = pb0 + wave * 32 + lane;
    float px = P[pid * 3 + 0], py = P[pid * 3 + 1], pz = P[pid * 3 + 2];
    int i0 = (int)ia[0];
    float cx = sxs[i0], cy = sys[i0], cz = szs[i0];
    float qx = px - cx, qy = py - cy, qz = pz - cz;
    float best = 3.0e38f;
#pragma unroll
    for (int j = 1; j < KNN; ++j) {
        int ij = (int)ia[j];
        float ex = sxs[ij] - cx, ey = sys[ij] - cy, ez = szs[ij] - cz;
        float el2 = ex * ex + ey * ey + ez * ez;
        float t   = qx * ex + qy * ey + qz * ez - 0.5f * el2;
        // (t/|e|)^2 == (dot - el2/2)^2 / el2 : avoids sqrt
        float sq  = (t * t) / el2;
        best = fminf(best, sq);
    }
    out[(size_t)b * NPTS + pid] = best;
}

extern "C" void kernel_launch(void* const* d_in, const int* in_sizes, int n_in,
                              void* d_out, int out_size, void* d_ws, size_t ws_size,
                              hipStream_t stream) {
    const float* points  = (const float*)d_in[0];   // (2, 16384, 3) f32
    const float* spoints = (const float*)d_in[1];   // (2, 4096, 3)  f32
    float* out = (float*)d_out;                     // (2, 16384)    f32
    (void)in_sizes; (void)n_in; (void)out_size; (void)d_ws; (void)ws_size;

    dim3 grid(2 * BPB);   // 128 blocks: 64 per batch
    dim3 block(BLK);      // 256 threads = 8 wave32s, 32 points per wave
    hipLaunchKernelGGL(voroloss_kernel, grid, block, 0, stream, points, spoints, out);
}